// Diagonal_80066780332682
// MI455X (gfx1250) — compile-verified
//
#include <hip/hip_runtime.h>

typedef __attribute__((ext_vector_type(4))) float v4f;

constexpr int B = 4096, I = 30000, O = 10000;
constexpr int THREADS = 256;
constexpr int OPT = 4;                  // outputs per thread
constexpr int O_TILE = THREADS * OPT;   // 1024 outputs per block tile
constexpr int X_TILE = 3 * O_TILE;      // 3072 floats = 12 KB per x row tile
constexpr int ROWS_PER_BLOCK = 16;
constexpr int O_TILES = (O + O_TILE - 1) / O_TILE;  // 10

__global__ __launch_bounds__(THREADS) void band_stream_kernel(
    const float* __restrict__ x, const float* __restrict__ weight,
    const float* __restrict__ bias, float* __restrict__ out) {
  __shared__ __align__(16) float shx[2][X_TILE];  // double-buffered x tile
  __shared__ __align__(16) float shw[X_TILE];     // weight band for tile
  __shared__ __align__(16) float shb[O_TILE];     // bias for tile

  const int tid = threadIdx.x;
  const int o_base = blockIdx.x * O_TILE;
  const int row0 = blockIdx.y * ROWS_PER_BLOCK;
  const int o0 = o_base + OPT * tid;
  const bool act = o0 < O;  // O % 4 == 0 -> all 4 outputs valid or none

  const unsigned voff = (unsigned)tid * 48u;  // byte offset of this thread's
                                              // 12 floats within the row tile

  // Async-copy this thread's 48 bytes of the row at `gbase` into LDS buffer
  // `buf`. GVS addressing: 64-bit SGPR base + 32-bit per-lane byte offset.
  // x is streamed exactly once -> non-temporal so the 491 MB stream does not
  // evict the L2-resident weight band shared by all row-sweeps.
  auto issue = [&](int buf, const float* gbase) {
    unsigned lds = (unsigned)(unsigned long long)&shx[buf][12 * tid];
    asm volatile(
        "global_load_async_to_lds_b128 %0, %1, %2 offset:0 th:TH_LOAD_NT\n\t"
        "global_load_async_to_lds_b128 %0, %1, %2 offset:16 th:TH_LOAD_NT\n\t"
        "global_load_async_to_lds_b128 %0, %1, %2 offset:32 th:TH_LOAD_NT"
        :
        : "v"(lds), "v"(voff), "s"(gbase)
        : "memory");
  };

  // Base of this block's x column window for its first row; advance by I
  // floats per batch row (uniform scalar add instead of a 64-bit mul).
  const float* xrow = x + (size_t)row0 * I + (size_t)3 * o_base;

  // Kick off the first row's x-tile before the weight-band gather so the
  // async DMA overlaps the (L2-resident) band fill.
  if (act) issue(0, xrow);

  // Gather diagonal band of `weight` + bias into LDS once per block;
  // reused across ROWS_PER_BLOCK rows.
  for (int k = tid; k < O_TILE; k += THREADS) {
    int o = o_base + k;
    if (o < O) {
      const float* wp = weight + (size_t)o * I + (size_t)3 * o;
      shw[3 * k + 0] = wp[0];
      shw[3 * k + 1] = wp[1];
      shw[3 * k + 2] = wp[2];
      shb[k] = bias[o];
    }
  }
  __syncthreads();

  v4f w0 = {}, w1 = {}, w2 = {}, bb = {};
  if (act) {
    const v4f* wv = (const v4f*)&shw[12 * tid];
    w0 = wv[0];
    w1 = wv[1];
    w2 = wv[2];
    bb = *(const v4f*)&shb[OPT * tid];
  }

  for (int r = 0; r < ROWS_PER_BLOCK; ++r) {
    if (r + 1 < ROWS_PER_BLOCK) {
      // Ensure prior DS reads of the buffer we are about to overwrite have
      // completed (async-engine LDS writes are unordered vs. our ds loads),
      // then prefetch the next row asynchronously.
      asm volatile("s_wait_dscnt 0x0" ::: "memory");
      if (act) issue((r + 1) & 1, xrow + (size_t)(r + 1) * I);
      // 3 async loads just issued may remain in flight; the previous 3
      // (current row) must be complete. Async loads retire in order.
      asm volatile("s_wait_asynccnt 0x3" ::: "memory");
    } else {
      asm volatile("s_wait_asynccnt 0x0" ::: "memory");
    }

    if (act) {
      const v4f* xs = (const v4f*)&shx[r & 1][12 * tid];
      v4f a = xs[0], b = xs[1], c = xs[2];
      v4f res;
      res.x = fmaf(a.x, w0.x, fmaf(a.y, w0.y, fmaf(a.z, w0.z, bb.x)));
      res.y = fmaf(a.w, w0.w, fmaf(b.x, w1.x, fmaf(b.y, w1.y, bb.y)));
      res.z = fmaf(b.z, w1.z, fmaf(b.w, w1.w, fmaf(c.x, w2.x, bb.z)));
      res.w = fmaf(c.y, w2.y, fmaf(c.z, w2.z, fmaf(c.w, w2.w, bb.w)));
      // Output is written once and never re-read: bypass caches.
      __builtin_nontemporal_store(res,
                                  (v4f*)(out + (size_t)(row0 + r) * O + o0));
    }
  }
}

extern "C" void kernel_launch(void* const* d_in, const int* in_sizes, int n_in,
                              void* d_out, int out_size, void* d_ws,
                              size_t ws_size, hipStream_t stream) {
  (void)in_sizes;
  (void)n_in;
  (void)d_ws;
  (void)ws_size;
  (void)out_size;
  const float* x = (const float*)d_in[0];
  const float* w = (const float*)d_in[1];
  const float* b = (const float*)d_in[2];
  float* out = (float*)d_out;
  dim3 grid(O_TILES, B / ROWS_PER_BLOCK);  // (10, 256)
  band_stream_kernel<<<grid, THREADS, 0, stream>>>(x, w, b, out);
}